// UResidualMambaChannel_68564857913693
// MI455X (gfx1250) — compile-verified
//
#include <hip/hip_runtime.h>
#include <hip/hip_bf16.h>

#define SEQ_L 1024
#define BATCH 8

typedef __attribute__((ext_vector_type(2))) float v2f;
typedef __attribute__((ext_vector_type(8))) float v8f;

// ============================================================================
// Generic conv1x1 / projection GEMM:  Y[b,o,l] = sum_k W[o,k] * X[b,k,l]
//   - X may be a channel-slice of a larger buffer (xStrideB = per-batch stride)
//   - optional elementwise second input (X := X * Xmul), same strides
//   - epilogue: +bias, activation (0=none,1=silu,2=softplus), +R1, +R2
// Block: 256 threads (8 wave32), tile M=64, N=64, K-step 32.
// LDS is row-major [m][k] / [n][k] (stride 36) so each WMMA fragment is one
// aligned ds_load_b64. Full tiles stage via global_load_b128.
// ============================================================================
#define TK   32
#define LDST 36   // LDS row stride in floats (even, multiple of 4)

__global__ __launch_bounds__(256) void gemm_wmma_kernel(
    const float* __restrict__ W, const float* __restrict__ bias,
    const float* __restrict__ X, const float* __restrict__ Xmul,
    long xStrideB, float* __restrict__ Y,
    const float* __restrict__ R1, const float* __restrict__ R2,
    int O, int K, int act)
{
    __shared__ float sW[64 * LDST];  // [m][k]
    __shared__ float sX[64 * LDST];  // [n][k]

    const int L = SEQ_L;
    const int tid  = threadIdx.x;
    const int lane = tid & 31;
    const int wave = tid >> 5;
    const int mw = wave & 3;   // 0..3 : M sub-tile
    const int nw = wave >> 2;  // 0..1 : N half

    const int o0 = blockIdx.x * 64;
    const int tilesPerBatch = L / 64;
    const int b  = blockIdx.y / tilesPerBatch;
    const int l0 = (blockIdx.y % tilesPerBatch) * 64;

    const float* Xb = X + (long)b * xStrideB + l0;
    const float* Xm = Xmul ? (Xmul + (long)b * xStrideB + l0) : nullptr;

    v8f acc[2] = {};
    const int mLane    = lane & 15;
    const int koffLane = (lane >> 4) << 1;  // lanes 0-15 -> K+0/+1, lanes 16-31 -> K+2/+3

    const bool kMul4   = ((K & 3) == 0);
    const bool oFull   = (o0 + 64 <= O);

    const float* swp  = &sW[(mw * 16 + mLane) * LDST];
    const float* sxp0 = &sX[(nw * 32 + mLane) * LDST];
    const float* sxp1 = &sX[(nw * 32 + 16 + mLane) * LDST];

    for (int kb = 0; kb < K; kb += TK) {
        const bool kFull = (kb + TK <= K);

        // ---- stage W tile: sW[m][k] = W[o0+m][kb+k] ----
        if (kFull && oFull && kMul4) {
            #pragma unroll
            for (int i = 0; i < 2; ++i) {
                int t  = tid + i * 256;       // 0..511
                int m  = t >> 3;              // 64 rows
                int k4 = (t & 7) << 2;        // 8 quads / row
                const float4 v = *(const float4*)(W + (long)(o0 + m) * K + kb + k4);
                *(float4*)(&sW[m * LDST + k4]) = v;
            }
        } else {
            #pragma unroll
            for (int i = 0; i < 8; ++i) {
                int idx = tid + i * 256;
                int k = idx & 31;
                int m = idx >> 5;
                int oo = o0 + m, kk = kb + k;
                sW[m * LDST + k] = (oo < O && kk < K) ? W[(long)oo * K + kk] : 0.0f;
            }
        }
        // ---- stage X tile: sX[n][k] = X[b][kb+k][l0+n]  (opt. * Xmul) ----
        if (kFull) {
            #pragma unroll
            for (int i = 0; i < 2; ++i) {
                int t  = tid + i * 256;       // 0..511
                int k  = t >> 4;              // 32 rows
                int n4 = (t & 15) << 2;       // 16 quads / row
                float4 v = *(const float4*)(Xb + (long)k * L + n4);
                if (Xm) {
                    const float4 m2 = *(const float4*)(Xm + (long)k * L + n4);
                    v.x *= m2.x; v.y *= m2.y; v.z *= m2.z; v.w *= m2.w;
                }
                sX[(n4 + 0) * LDST + k] = v.x;
                sX[(n4 + 1) * LDST + k] = v.y;
                sX[(n4 + 2) * LDST + k] = v.z;
                sX[(n4 + 3) * LDST + k] = v.w;
            }
        } else {
            #pragma unroll
            for (int i = 0; i < 8; ++i) {
                int idx = tid + i * 256;
                int n = idx & 63;
                int k = idx >> 6;
                int kk = kb + k;
                float v = 0.0f;
                if (kk < K) {
                    v = Xb[(long)kk * L + n];
                    if (Xm) v *= Xm[(long)kk * L + n];
                }
                sX[n * LDST + k] = v;
            }
        }
        __syncthreads();

        #pragma unroll
        for (int kk4 = 0; kk4 < TK / 4; ++kk4) {
            const int kbase = kk4 * 4 + koffLane;               // even -> b64 aligned
            const v2f a  = *(const v2f*)(swp  + kbase);
            const v2f b0 = *(const v2f*)(sxp0 + kbase);
            const v2f b1 = *(const v2f*)(sxp1 + kbase);
            acc[0] = __builtin_amdgcn_wmma_f32_16x16x4_f32(false, a, false, b0, (short)0, acc[0], false, false);
            acc[1] = __builtin_amdgcn_wmma_f32_16x16x4_f32(false, a, false, b1, (short)0, acc[1], false, false);
        }
        __syncthreads();
    }

    // epilogue + store. C/D layout: vgpr v -> M = v + 8*(lane>=16), N = lane&15
    const long ybase = (long)b * O * L;
    const int mHi = (lane >> 4) << 3;
    #pragma unroll
    for (int ns = 0; ns < 2; ++ns) {
        int l = l0 + nw * 32 + ns * 16 + mLane;
        #pragma unroll
        for (int v = 0; v < 8; ++v) {
            int o = o0 + mw * 16 + mHi + v;
            if (o < O) {
                float val = acc[ns][v];
                if (bias) val += bias[o];
                if (act == 1) {                       // silu
                    val = val / (1.0f + __expf(-val));
                } else if (act == 2) {                // softplus
                    val = (val > 20.0f) ? val : log1pf(__expf(val));
                }
                long yi = ybase + (long)o * L + l;
                if (R1) val += R1[yi];
                if (R2) val += R2[yi];
                Y[yi] = val;
            }
        }
    }
}

// ============================================================================
// RMSNorm over channel dim of (B,C,L), per position (b,l).
// ============================================================================
__global__ void rmsnorm_kernel(const float* __restrict__ x, float* __restrict__ y, int C)
{
    const int L = SEQ_L;
    int idx = blockIdx.x * blockDim.x + threadIdx.x;  // B*L threads
    int b = idx >> 10, l = idx & (L - 1);
    long base = (long)b * C * L + l;
    float ss = 0.0f;
    for (int c = 0; c < C; ++c) { float v = x[base + (long)c * L]; ss += v * v; }
    float r = rsqrtf(ss / (float)C + 1e-5f);
    for (int c = 0; c < C; ++c) y[base + (long)c * L] = x[base + (long)c * L] * r;
}

// s[b,c,l] = rmsnorm(m)[b,c,l] + s[b,c,l]
__global__ void rmsnorm_add_kernel(const float* __restrict__ m, float* __restrict__ s, int C)
{
    const int L = SEQ_L;
    int idx = blockIdx.x * blockDim.x + threadIdx.x;
    int b = idx >> 10, l = idx & (L - 1);
    long base = (long)b * C * L + l;
    float ss = 0.0f;
    for (int c = 0; c < C; ++c) { float v = m[base + (long)c * L]; ss += v * v; }
    float r = rsqrtf(ss / (float)C + 1e-5f);
    for (int c = 0; c < C; ++c) s[base + (long)c * L] += m[base + (long)c * L] * r;
}

// ============================================================================
// Depthwise causal conv (width 4) + bias + SiLU. Input = xi half of xz buffer.
// ============================================================================
__global__ void dwconv_kernel(const float* __restrict__ xz, const float* __restrict__ cw,
                              const float* __restrict__ cb, float* __restrict__ xi, int Di)
{
    const int L = SEQ_L;
    long idx = (long)blockIdx.x * blockDim.x + threadIdx.x;
    long total = (long)BATCH * Di * L;
    if (idx >= total) return;
    int l = (int)(idx & (L - 1));
    long t = idx >> 10;
    int d = (int)(t % Di);
    int b = (int)(t / Di);
    const float* xp = xz + ((long)b * 2 * Di + d) * L;  // xi slice = channels [0,Di)
    const float* w = cw + d * 4;                        // conv_w (Di,1,4)
    float a = cb[d];
    #pragma unroll
    for (int j = 0; j < 4; ++j) {
        int li = l - 3 + j;
        if (li >= 0) a += w[j] * xp[li];
    }
    a = a / (1.0f + __expf(-a));  // silu
    xi[((long)b * Di + d) * L + l] = a;
}

// ============================================================================
// Selective scan: one thread per (b,d), 16 states in registers, serial in L.
// Fuses  y = scan + xi*D  and  y *= silu(z).
// ============================================================================
__global__ void scan_kernel(
    const float* __restrict__ A_log, const float* __restrict__ Dp,
    const float* __restrict__ dt, const float* __restrict__ xi,
    const float* __restrict__ z, long zStrideB,
    const float* __restrict__ Bm, const float* __restrict__ Cm, long dbcStrideB,
    float* __restrict__ ym, int Di)
{
    const int L = SEQ_L;
    int d = blockIdx.x * blockDim.x + threadIdx.x;
    int b = blockIdx.y;
    if (d >= Di) return;

    float An[16], h[16];
    #pragma unroll
    for (int n = 0; n < 16; ++n) { An[n] = -__expf(A_log[d * 16 + n]); h[n] = 0.0f; }
    float Dd = Dp[d];

    const float* dtp = dt + ((long)b * Di + d) * L;
    const float* xip = xi + ((long)b * Di + d) * L;
    const float* zp  = z  + (long)b * zStrideB + (long)d * L;
    const float* Bp  = Bm + (long)b * dbcStrideB;
    const float* Cp  = Cm + (long)b * dbcStrideB;
    float* yp = ym + ((long)b * Di + d) * L;

    for (int l = 0; l < L; ++l) {
        float dtv = dtp[l];
        float xv  = xip[l];
        float acc = 0.0f;
        #pragma unroll
        for (int n = 0; n < 16; ++n) {
            float bn = Bp[(long)n * L + l];
            float cn = Cp[(long)n * L + l];
            h[n] = h[n] * __expf(dtv * An[n]) + (dtv * bn) * xv;
            acc += h[n] * cn;
        }
        float zv = zp[l];
        float y = acc + xv * Dd;
        y *= zv / (1.0f + __expf(-zv));  // * silu(z)
        yp[l] = y;
    }
}

// ============================================================================
// Host-side orchestration
// ============================================================================
namespace {

struct Cur {
    void* const* din;
    int i;
    const float* next() { return (const float*)din[i++]; }
};

struct EncP {
    const float *ip_w, *ip_b, *fp_w, *fp_b, *wp_w, *wp_b, *op_w, *op_b;
    const float *in_w, *conv_w, *conv_b, *x_w, *dt_w, *dt_b, *A_log, *Dvec, *out_w;
};

EncP readEnc(Cur& c) {
    EncP e;
    e.ip_w = c.next(); e.ip_b = c.next();
    e.fp_w = c.next(); e.fp_b = c.next();
    e.wp_w = c.next(); e.wp_b = c.next();
    e.op_w = c.next(); e.op_b = c.next();
    e.in_w = c.next(); e.conv_w = c.next(); e.conv_b = c.next();
    e.x_w = c.next(); e.dt_w = c.next(); e.dt_b = c.next();
    e.A_log = c.next(); e.Dvec = c.next(); e.out_w = c.next();
    return e;
}

struct Bufs {
    float *X0, *X1, *RES, *H, *T, *S, *XZ, *XI, *DBC, *DT, *YM;
};

void gemm(hipStream_t st, const float* W, const float* bias, const float* X,
          const float* Xmul, long xStrideB, float* Y, const float* R1,
          const float* R2, int O, int K, int act)
{
    dim3 g((O + 63) / 64, BATCH * (SEQ_L / 64));
    gemm_wmma_kernel<<<g, 256, 0, st>>>(W, bias, X, Xmul, xStrideB, Y, R1, R2, O, K, act);
}

// Encoder at channel count c: xin (B,c,L) -> xout (B,c,L), optional extra skip res2.
void run_encoder(hipStream_t st, const Bufs& w, const EncP& p, int c,
                 const float* xin, float* xout, const float* res2)
{
    const int L = SEQ_L, B = BATCH;
    const int d = 2 * c, Di = 4 * c;
    const int r = (2 * c + 15) / 16;
    const int posBlocks = (B * L) / 256;

    // 1. h = rmsnorm(x)
    rmsnorm_kernel<<<posBlocks, 256, 0, st>>>(xin, w.H, c);
    // 2. t1 = ip_w @ h + ip_b
    gemm(st, p.ip_w, p.ip_b, w.H, nullptr, (long)c * L, w.T, nullptr, nullptr, 2 * c, c, 0);
    // 3. s0 = fp_w @ t1 + fp_b
    gemm(st, p.fp_w, p.fp_b, w.T, nullptr, (long)2 * c * L, w.S, nullptr, nullptr, 2 * c, 2 * c, 0);
    // 4. xz = in_w @ s0                       (mamba in-proj, O=2Di, K=d)
    gemm(st, p.in_w, nullptr, w.S, nullptr, (long)d * L, w.XZ, nullptr, nullptr, 2 * Di, d, 0);
    // 5. xi = silu(causal dwconv4(xz[:Di]) + conv_b)
    {
        long tot = (long)B * Di * L;
        dwconv_kernel<<<(unsigned)((tot + 255) / 256), 256, 0, st>>>(w.XZ, p.conv_w, p.conv_b, w.XI, Di);
    }
    // 6. dbc = x_w @ xi                       (O=r+32, K=Di)
    gemm(st, p.x_w, nullptr, w.XI, nullptr, (long)Di * L, w.DBC, nullptr, nullptr, r + 32, Di, 0);
    // 7. dt = softplus(dt_w @ dbc[:r] + dt_b) (O=Di, K=r)
    gemm(st, p.dt_w, p.dt_b, w.DBC, nullptr, (long)(r + 32) * L, w.DT, nullptr, nullptr, Di, r, 2);
    // 8. selective scan (fused +xi*D, *silu(z))
    {
        dim3 sg((Di + 127) / 128, B);
        scan_kernel<<<sg, 128, 0, st>>>(p.A_log, p.Dvec, w.DT, w.XI,
                                        w.XZ + (long)Di * L, (long)2 * Di * L,
                                        w.DBC + (long)r * L, w.DBC + (long)(r + 16) * L,
                                        (long)(r + 32) * L, w.YM, Di);
    }
    // 9. mo = out_w @ ym  (reuse T; t1 dead)
    gemm(st, p.out_w, nullptr, w.YM, nullptr, (long)Di * L, w.T, nullptr, nullptr, d, Di, 0);
    // 10. s = rmsnorm(mo) + s0
    rmsnorm_add_kernel<<<posBlocks, 256, 0, st>>>(w.T, w.S, 2 * c);
    // 11. gate = silu(wp_w @ h + wp_b)  (reuse DT; dead after scan)
    gemm(st, p.wp_w, p.wp_b, w.H, nullptr, (long)c * L, w.DT, nullptr, nullptr, 2 * c, c, 1);
    // 12. out = op_w @ (gate * s) + op_b + xin [+ res2]
    gemm(st, p.op_w, p.op_b, w.DT, w.S, (long)2 * c * L, xout, xin, res2, c, 2 * c, 0);
}

} // namespace

extern "C" void kernel_launch(void* const* d_in, const int* in_sizes, int n_in,
                              void* d_out, int out_size, void* d_ws, size_t ws_size,
                              hipStream_t stream)
{
    const float* x0 = (const float*)d_in[0];
    Cur cur{d_in, 1};

    // Workspace carve-up (all fp32 element counts; maxima over the net)
    float* ws = (float*)d_ws;
    size_t off = 0;
    auto take = [&](size_t n) { float* p = ws + off; off += n; return p; };
    Bufs w;
    w.X0  = take(2097152);   // (B,256,L)
    w.X1  = take(2097152);
    w.RES = take(1048576 + 524288 + 262144);  // skip stack for levels 1..3
    w.H   = take(2097152);   // (B,c,L),  c<=256
    w.T   = take(4194304);   // (B,2c,L)
    w.S   = take(4194304);
    w.XZ  = take(16777216);  // (B,2Di,L), Di<=1024
    w.XI  = take(8388608);   // (B,Di,L)
    w.DBC = take(524288);    // (B,r+32,L), r+32<=64
    w.DT  = take(8388608);
    w.YM  = take(8388608);

    const float* res[4];
    float* resSlab = w.RES;
    size_t resOff = 0;

    // ---- down path ----
    const float* curx = x0;
    for (int i = 0; i < 4; ++i) {
        int ci = 256 >> i, co = ci >> 1;
        const float* dw = cur.next();
        const float* db = cur.next();
        EncP e = readEnc(cur);
        if (i == 0) {
            res[0] = x0;  // input buffer is never mutated
        } else {
            size_t n = (size_t)BATCH * ci * SEQ_L;
            hipMemcpyAsync(resSlab + resOff, curx, n * sizeof(float),
                           hipMemcpyDeviceToDevice, stream);
            res[i] = resSlab + resOff;
            resOff += n;
        }
        gemm(stream, dw, db, curx, nullptr, (long)ci * SEQ_L, w.X1, nullptr, nullptr, co, ci, 0);
        run_encoder(stream, w, e, co, w.X1, w.X0, nullptr);
        curx = w.X0;
    }

    // ---- mids (c = 16) ----
    for (int m = 0; m < 2; ++m) {
        EncP e = readEnc(cur);
        float* outb = (curx == w.X0) ? w.X1 : w.X0;
        run_encoder(stream, w, e, 16, curx, outb, nullptr);
        curx = outb;
    }

    // ---- up path ----
    for (int j = 0; j < 4; ++j) {
        int i = 4 - j;
        int ci = 256 >> i;
        int c2 = 2 * ci;
        const float* uw = cur.next();
        const float* ub = cur.next();
        EncP e = readEnc(cur);
        float* t = (curx == w.X0) ? w.X1 : w.X0;
        gemm(stream, uw, ub, curx, nullptr, (long)ci * SEQ_L, t, nullptr, nullptr, c2, ci, 0);
        float* outb = (t == w.X0) ? w.X1 : w.X0;
        float* dst = (j == 3) ? (float*)d_out : outb;
        run_encoder(stream, w, e, c2, t, dst, res[3 - j]);
        curx = dst;
    }
    (void)in_sizes; (void)n_in; (void)out_size; (void)ws_size;
}